// HungarianMatcher_7344394076633
// MI455X (gfx1250) — compile-verified
//
#include <hip/hip_runtime.h>
#include <math.h>

#define BS 16
#define NQ 1024
#define NC 256
#define SL 1024
#define NT 128
#define NROW (BS*NQ)   // 16384 rows

typedef __attribute__((ext_vector_type(2))) float v2f;
typedef __attribute__((ext_vector_type(8))) float v8f;

// ---------------------------------------------------------------------------
// Kernel 1: per-row argmax over SL=1024 for pred_left (rows 0..16383) and
// pred_right (rows 16384..32767). One wave (32 lanes) per row, float4 loads.
// Tie-break: first (lowest) index, matching jnp.argmax.
// ---------------------------------------------------------------------------
__global__ void argmax_rows(const float* __restrict__ left,
                            const float* __restrict__ right,
                            int* __restrict__ al, int* __restrict__ ar) {
  const int wave = threadIdx.x >> 5, lane = threadIdx.x & 31;
  const int row = blockIdx.x * 8 + wave;            // 0..32767
  const bool is_left = (row < NROW);
  const int r = is_left ? row : row - NROW;
  const float* src = is_left ? left : right;
  const float4* p = (const float4*)(src + (size_t)r * SL);

  float bv = -INFINITY; int bi = 0;
#pragma unroll
  for (int i = 0; i < SL / 128; ++i) {              // 8 iterations, coalesced
    float4 x = p[i * 32 + lane];
    int c0 = (i * 32 + lane) * 4;
    if (x.x > bv) { bv = x.x; bi = c0;     }
    if (x.y > bv) { bv = x.y; bi = c0 + 1; }
    if (x.z > bv) { bv = x.z; bi = c0 + 2; }
    if (x.w > bv) { bv = x.w; bi = c0 + 3; }
  }
  // wave32 butterfly reduce: max value, lowest index on tie
  for (int m = 16; m; m >>= 1) {
    float ov = __shfl_xor(bv, m, 32);
    int   oi = __shfl_xor(bi, m, 32);
    if (ov > bv || (ov == bv && oi < bi)) { bv = ov; bi = oi; }
  }
  if (lane == 0) {
    if (is_left) al[r] = bi; else ar[r] = bi;
  }
}

// ---------------------------------------------------------------------------
// Kernel 2: softmax row stats over NC=256 per row: rowmax m and 1/sum(exp(x-m)).
// One wave per row; each lane holds 8 floats in registers.
// ---------------------------------------------------------------------------
__global__ void softmax_stats(const float* __restrict__ logits,
                              float* __restrict__ mx, float* __restrict__ invz) {
  const int wave = threadIdx.x >> 5, lane = threadIdx.x & 31;
  const int row = blockIdx.x * 8 + wave;            // 0..16383
  const float4* p = (const float4*)(logits + (size_t)row * NC);
  float4 r0 = p[lane];
  float4 r1 = p[32 + lane];
  float m = fmaxf(fmaxf(fmaxf(r0.x, r0.y), fmaxf(r0.z, r0.w)),
                  fmaxf(fmaxf(r1.x, r1.y), fmaxf(r1.z, r1.w)));
  for (int s = 16; s; s >>= 1) m = fmaxf(m, __shfl_xor(m, s, 32));
  float z = __expf(r0.x - m) + __expf(r0.y - m) + __expf(r0.z - m) + __expf(r0.w - m)
          + __expf(r1.x - m) + __expf(r1.y - m) + __expf(r1.z - m) + __expf(r1.w - m);
  for (int s = 16; s; s >>= 1) z += __shfl_xor(z, s, 32);
  if (lane == 0) { mx[row] = m; invz[row] = 1.0f / z; }
}

// ---------------------------------------------------------------------------
// Kernel 3: cost matrix via one-hot GEMM on WMMA.
//   cost_class[q,t] = softmax(logits[q,:]) x (-onehot(labels[t]))
// computed with V_WMMA_F32_16X16X4_F32 (64 K-steps of 4 per 16x16 tile).
// Span cost |al-gl|+|ar-gr| is added to the accumulator, result written to
// d_out (Cb, row-major [b,q,t]) and as -C transposed ([b,t,q]) for the auction.
// One wave per 16x16 tile; 8192 tiles total.
// ---------------------------------------------------------------------------
__global__ void cost_wmma(const float* __restrict__ logits,
                          const int* __restrict__ labels,
                          const int* __restrict__ gtl,
                          const int* __restrict__ gtr,
                          const int* __restrict__ al,
                          const int* __restrict__ ar,
                          const float* __restrict__ mx,
                          const float* __restrict__ invz,
                          float* __restrict__ Cb,    // [BS*NQ*NT]
                          float* __restrict__ Xt) {  // [BS*NT*NQ] = -C^T
  const int wave = threadIdx.x >> 5, lane = threadIdx.x & 31;
  const int tile = blockIdx.x * 8 + wave;           // 0..8191
  const int tt = tile & 7;                          // t-tile (0..7)
  const int qt = (tile >> 3) & 63;                  // q-tile (0..63)
  const int b  = tile >> 9;                         // batch (0..15)
  const int l16 = lane & 15, hi = lane >> 4;

  // B-side: one column t per lane (same for both lane halves)
  const int t   = tt * 16 + l16;
  const int gti = b * NT + t;
  const int lab = labels[gti];
  const int gl  = gtl[gti];
  const int gr  = gtr[gti];

  // A-side: one row q per lane16; lanes 16-31 supply K+2,K+3 per ISA layout
  const int qrow = b * NQ + qt * 16 + l16;
  const float2* lp = (const float2*)(logits + (size_t)qrow * NC);
  const float mv = mx[qrow], iz = invz[qrow];

  v8f acc = {0.f, 0.f, 0.f, 0.f, 0.f, 0.f, 0.f, 0.f};
#pragma unroll 8
  for (int k0 = 0; k0 < NC; k0 += 4) {
    const int kk = k0 + 2 * hi;
    float2 x = lp[(k0 >> 1) + hi];
    v2f a;  a.x = __expf(x.x - mv) * iz;  a.y = __expf(x.y - mv) * iz;
    v2f bm; bm.x = (lab == kk)     ? -1.0f : 0.0f;
            bm.y = (lab == kk + 1) ? -1.0f : 0.0f;
    acc = __builtin_amdgcn_wmma_f32_16x16x4_f32(
        /*neg_a=*/false, a, /*neg_b=*/false, bm,
        /*c_mod=*/(short)0, acc, /*reuse_a=*/false, /*reuse_b=*/false);
  }

  // C/D layout: VGPR v -> row m = v + 8*hi, col n = lane16
#pragma unroll
  for (int v = 0; v < 8; ++v) {
    const int q  = qt * 16 + v + 8 * hi;
    const int qg = b * NQ + q;
    int d1 = al[qg] - gl; d1 = d1 < 0 ? -d1 : d1;
    int d2 = ar[qg] - gr; d2 = d2 < 0 ? -d2 : d2;
    const float cst = (float)(d1 + d2) + acc[v];
    Cb[(size_t)qg * NT + t] = cst;
    Xt[(size_t)b * (NT * NQ) + (size_t)t * NQ + q] = -cst;
  }
}

// ---------------------------------------------------------------------------
// Kernel 4: auction assignment, one workgroup per batch (exact reference
// semantics). Rows = targets (128) bid for columns = queries (1024).
// Bids resolved with packed-u64 LDS atomicMax: (f32 bid)<<32 | ~row, so the
// max picks the highest bid and the lowest row index on ties (jnp.argmax).
// Steal phase and assign phase are barrier-separated to match the reference's
// unassign-then-assign order.
// ---------------------------------------------------------------------------
__global__ void auction_kernel(const float* __restrict__ Xt,
                               float* __restrict__ out_pred,
                               float* __restrict__ out_tgt) {
  __shared__ float costS[NQ];
  __shared__ unsigned long long bidS[NQ];
  __shared__ int ownerS[NQ];
  __shared__ int assS[NT];
  __shared__ int nunS;

  const int b = blockIdx.x;
  const int tid = threadIdx.x;                      // 256 threads = 8 waves
  const int wave = tid >> 5, lane = tid & 31;
  const float* X = Xt + (size_t)b * (NT * NQ);

  for (int c = tid; c < NQ; c += 256) { costS[c] = 0.f; ownerS[c] = -1; }
  if (tid < NT) assS[tid] = -1;
  __syncthreads();

  const float eps = 1.0f / (float)NT;
  for (int it = 0; it < 100000; ++it) {
    for (int c = tid; c < NQ; c += 256) bidS[c] = 0ull;
    if (tid == 0) nunS = 0;
    __syncthreads();

    // --- bidding: only unassigned rows (assigned rows have inc==0) ---
    for (int r = wave; r < NT; r += 8) {
      if (assS[r] >= 0) continue;
      const float* xr = X + (size_t)r * NQ;
      float v1 = -INFINITY, v2 = -INFINITY; int i1 = 0;
      for (int c = lane; c < NQ; c += 32) {
        float val = xr[c] - costS[c];
        if (val > v1) { v2 = v1; v1 = val; i1 = c; }
        else if (val > v2) { v2 = val; }
      }
      for (int m = 16; m; m >>= 1) {
        float ov1 = __shfl_xor(v1, m, 32);
        float ov2 = __shfl_xor(v2, m, 32);
        int   oi1 = __shfl_xor(i1, m, 32);
        if (ov1 > v1 || (ov1 == v1 && oi1 < i1)) {
          v2 = fmaxf(v1, ov2); v1 = ov1; i1 = oi1;
        } else {
          v2 = fmaxf(v2, ov1);
        }
      }
      if (lane == 0) {
        const float inc = v1 - v2 + eps;            // > 0 always
        const unsigned long long pk =
            ((unsigned long long)__float_as_uint(inc) << 32) |
            (unsigned long long)(0xFFFFFFFFu - (unsigned)r);
        atomicMax(&bidS[i1], pk);
      }
    }
    __syncthreads();

    // --- steal: unassign previous owners of columns that received bids ---
    for (int c = tid; c < NQ; c += 256) {
      if (bidS[c]) { int pr = ownerS[c]; if (pr >= 0) assS[pr] = -1; }
    }
    __syncthreads();

    // --- assign winners, raise prices ---
    for (int c = tid; c < NQ; c += 256) {
      unsigned long long bv = bidS[c];
      if (bv) {
        costS[c] += __uint_as_float((unsigned)(bv >> 32));
        int w = (int)(0xFFFFFFFFu - (unsigned)bv);
        ownerS[c] = w;
        assS[w] = c;
      }
    }
    __syncthreads();

    if (tid < NT && assS[tid] < 0) atomicAdd(&nunS, 1);
    __syncthreads();
    int nun = nunS;
    __syncthreads();                                // protect nunS reset
    if (nun == 0) break;
  }

  if (tid < NT) {
    out_pred[b * NT + tid] = (float)assS[tid];
    out_tgt[b * NT + tid]  = (float)tid;
  }
}

// ---------------------------------------------------------------------------
extern "C" void kernel_launch(void* const* d_in, const int* in_sizes, int n_in,
                              void* d_out, int out_size, void* d_ws, size_t ws_size,
                              hipStream_t stream) {
  const float* pred_logits = (const float*)d_in[0];
  const float* pred_left   = (const float*)d_in[1];
  const float* pred_right  = (const float*)d_in[2];
  const int*   labels      = (const int*)d_in[3];
  const int*   gt_left     = (const int*)d_in[4];
  const int*   gt_right    = (const int*)d_in[5];

  float* out      = (float*)d_out;
  float* out_pred = out;                 // [BS*NT]
  float* out_tgt  = out + BS * NT;       // [BS*NT]
  float* Cb       = out + 2 * BS * NT;   // [BS*NQ*NT]

  // workspace layout (floats/ints): al, ar, mx, invz, Xt  (~8.7 MB)
  int*   al   = (int*)d_ws;
  int*   ar   = al + NROW;
  float* mx   = (float*)(ar + NROW);
  float* invz = mx + NROW;
  float* Xt   = invz + NROW;             // [BS*NT*NQ]

  argmax_rows  <<<(2 * NROW) / 8, 256, 0, stream>>>(pred_left, pred_right, al, ar);
  softmax_stats<<<NROW / 8,       256, 0, stream>>>(pred_logits, mx, invz);
  cost_wmma    <<<(BS * 64 * 8) / 8, 256, 0, stream>>>(pred_logits, labels, gt_left,
                                                       gt_right, al, ar, mx, invz,
                                                       Cb, Xt);
  auction_kernel<<<BS, 256, 0, stream>>>(Xt, out_pred, out_tgt);
}